// DMTetGeometry_12524124635504
// MI455X (gfx1250) — compile-verified
//
#include <hip/hip_runtime.h>
#include <stdint.h>

// =====================================================================
// DMTet marching-tetrahedra for MI455X (gfx1250, wave32).
// Memory-bound irregular pipeline; no GEMM -> no WMMA. CDNA5 paths used:
//   - global_load_async_to_lds_b128 + s_wait_asynccnt (tet row staging)
//   - wave32 ballot/shfl scans (1-2 barriers per block scan vs 16)
//   - parallel per-digit histogram scan (radix sort no longer serialized)
// All data-dependent sizes (T, NF1, NF2, M) live in a device counter
// block; grids are worst-case sized and bound-check on device, so the
// whole pipeline is a fixed, graph-capturable kernel sequence.
// =====================================================================

#define WGS 256
#define NWAVE 8              // 256 / wave32

__constant__ int c_tri[16][6] = {
 {-1,-1,-1,-1,-1,-1},{1,0,2,-1,-1,-1},{4,0,3,-1,-1,-1},{1,4,2,1,3,4},
 {3,1,5,-1,-1,-1},{2,3,0,2,5,3},{1,4,0,1,5,4},{4,2,5,-1,-1,-1},
 {4,5,2,-1,-1,-1},{4,1,0,4,5,1},{3,2,0,3,5,2},{1,3,5,-1,-1,-1},
 {4,1,2,4,3,1},{3,0,4,-1,-1,-1},{2,0,1,-1,-1,-1},{-1,-1,-1,-1,-1,-1}};
__constant__ int c_ntri[16] = {0,1,1,2,1,2,2,1,1,2,2,1,2,1,1,0};
__constant__ int c_edge[12] = {0,1,0,2,0,3,1,2,1,3,2,3};

// ---- wave32 primitives ------------------------------------------------
__device__ __forceinline__ unsigned wave_incl_scan_u32(unsigned v) {
  unsigned lane = threadIdx.x & 31u;
  #pragma unroll
  for (int off = 1; off < 32; off <<= 1) {
    unsigned t = __shfl_up(v, off, 32);
    if (lane >= (unsigned)off) v += t;
  }
  return v;
}

// block-wide exclusive scan of general u32 (all 256 threads must call).
// wsum must be 8 u32 of LDS. blockTotal returned to every thread.
__device__ __forceinline__ unsigned block_excl_scan_u32(unsigned v, unsigned* wsum,
                                                        unsigned& blockTotal) {
  unsigned lane = threadIdx.x & 31u, wid = threadIdx.x >> 5;
  unsigned incl = wave_incl_scan_u32(v);
  if (lane == 31u) wsum[wid] = incl;
  __syncthreads();
  unsigned base = 0u, tot = 0u;
  #pragma unroll
  for (unsigned w = 0; w < NWAVE; w++) {
    unsigned s = wsum[w];
    if (w < wid) base += s;
    tot += s;
  }
  blockTotal = tot;
  __syncthreads();           // wsum safe to reuse after return
  return base + incl - v;
}

// block-wide exclusive scan of a 0/1 flag via ballot (all threads call).
__device__ __forceinline__ unsigned block_excl_scan_flag(unsigned flag, unsigned* wsum) {
  unsigned lane = threadIdx.x & 31u, wid = threadIdx.x >> 5;
  unsigned m = (unsigned)__ballot(flag != 0u);
  unsigned pre = __popc(m & ((1u << lane) - 1u));
  if (lane == 0u) wsum[wid] = (unsigned)__popc(m);
  __syncthreads();
  unsigned base = 0u;
  #pragma unroll
  for (unsigned w = 0; w < NWAVE; w++)
    if (w < wid) base += wsum[w];
  return base + pre;
}

// ------------------------------------------------------------------
// 1) classify: per tet -> 4-bit occupancy case; per-block flag counts.
//    Tet rows staged to LDS with CDNA5 async global->LDS copies.
// ------------------------------------------------------------------
__global__ void k_classify(const int4* __restrict__ tet, const float* __restrict__ sdf,
                           int F, unsigned* blkV, unsigned* blk1, unsigned* blk2,
                           unsigned char* caseArr) {
  __shared__ int4 stage[WGS];
  __shared__ unsigned cnt[3];
  int t = blockIdx.x * WGS + threadIdx.x;
  if (threadIdx.x < 3) cnt[threadIdx.x] = 0u;

#if defined(__HIP_DEVICE_COMPILE__)
  if (t < F) {
    unsigned lds_off = (unsigned)(uintptr_t)(&stage[threadIdx.x]); // LDS offset in low 32 bits
    unsigned voff    = (unsigned)t * 16u;                          // byte offset into tet array
    asm volatile("global_load_async_to_lds_b128 %0, %1, %2"
                 :: "v"(lds_off), "v"(voff), "s"(tet) : "memory");
  }
  asm volatile("s_wait_asynccnt 0" ::: "memory");
#else
  if (t < F) stage[threadIdx.x] = tet[t];
#endif
  __syncthreads();

  unsigned fV = 0u, f1 = 0u, f2 = 0u;
  if (t < F) {
    int4 r = stage[threadIdx.x];
    unsigned cs = (sdf[r.x] > 0.f ? 1u : 0u) | (sdf[r.y] > 0.f ? 2u : 0u) |
                  (sdf[r.z] > 0.f ? 4u : 0u) | (sdf[r.w] > 0.f ? 8u : 0u);
    caseArr[t] = (unsigned char)cs;
    int nt = c_ntri[cs];
    fV = (nt > 0);  f1 = (nt == 1);  f2 = (nt == 2);
  }
  // per-wave ballot reduce, one LDS atomic per wave per flag
  unsigned mV = (unsigned)__ballot(fV != 0u);
  unsigned m1 = (unsigned)__ballot(f1 != 0u);
  unsigned m2 = (unsigned)__ballot(f2 != 0u);
  if ((threadIdx.x & 31u) == 0u) {
    if (mV) atomicAdd(&cnt[0], (unsigned)__popc(mV));
    if (m1) atomicAdd(&cnt[1], (unsigned)__popc(m1));
    if (m2) atomicAdd(&cnt[2], (unsigned)__popc(m2));
  }
  __syncthreads();
  if (threadIdx.x == 0) {
    blkV[blockIdx.x] = cnt[0];
    blk1[blockIdx.x] = cnt[1];
    blk2[blockIdx.x] = cnt[2];
  }
}

// ------------------------------------------------------------------
// single-block exclusive scan (in place) of an array, total -> *totalOut
// (used for the small per-block-sum arrays only)
// ------------------------------------------------------------------
__global__ void k_scan_arr(unsigned* arr, int n, unsigned* totalOut) {
  __shared__ unsigned wsum[NWAVE];
  __shared__ unsigned carry;
  if (threadIdx.x == 0) carry = 0u;
  __syncthreads();
  for (int base = 0; base < n; base += WGS) {
    int i = base + threadIdx.x;
    unsigned v = (i < n) ? arr[i] : 0u;
    unsigned tot;
    unsigned ex = block_excl_scan_u32(v, wsum, tot);
    if (i < n) arr[i] = carry + ex;
    __syncthreads();
    if (threadIdx.x == 0) carry += tot;
    __syncthreads();
  }
  if (threadIdx.x == 0 && totalOut) *totalOut = carry;
}

// ------------------------------------------------------------------
// 2) compact valid tets (tet order) and 1-tri / 2-tri lists.
// ------------------------------------------------------------------
__global__ void k_compact(const unsigned char* __restrict__ caseArr, int F,
                          const unsigned* blkV, const unsigned* blk1, const unsigned* blk2,
                          unsigned* validIdx, unsigned char* validCase,
                          unsigned* tri1, unsigned* tri2) {
  __shared__ unsigned wsum[3 * NWAVE];
  int t = blockIdx.x * WGS + threadIdx.x;
  unsigned cs = (t < F) ? (unsigned)caseArr[t] : 0u;
  int nt = (t < F) ? c_ntri[cs] : 0;
  unsigned fV = (nt > 0) ? 1u : 0u;
  unsigned f1 = (nt == 1) ? 1u : 0u;
  unsigned f2 = (nt == 2) ? 1u : 0u;

  unsigned lane = threadIdx.x & 31u, wid = threadIdx.x >> 5;
  unsigned mV = (unsigned)__ballot(fV != 0u);
  unsigned m1 = (unsigned)__ballot(f1 != 0u);
  unsigned m2 = (unsigned)__ballot(f2 != 0u);
  unsigned lm = (1u << lane) - 1u;
  unsigned pV = __popc(mV & lm), p1 = __popc(m1 & lm), p2 = __popc(m2 & lm);
  if (lane == 0u) {
    wsum[wid]             = (unsigned)__popc(mV);
    wsum[NWAVE + wid]     = (unsigned)__popc(m1);
    wsum[2 * NWAVE + wid] = (unsigned)__popc(m2);
  }
  __syncthreads();
  unsigned bV = 0u, b1 = 0u, b2 = 0u;
  #pragma unroll
  for (unsigned w = 0; w < NWAVE; w++)
    if (w < wid) { bV += wsum[w]; b1 += wsum[NWAVE + w]; b2 += wsum[2 * NWAVE + w]; }

  if (fV) {
    unsigned rv = blkV[blockIdx.x] + bV + pV;
    validIdx[rv]  = (unsigned)t;
    validCase[rv] = (unsigned char)cs;
    if (f1) tri1[blk1[blockIdx.x] + b1 + p1] = rv;
    else    tri2[blk2[blockIdx.x] + b2 + p2] = rv;
  }
}

// ------------------------------------------------------------------
// 3) generate packed edge items: item = key<<24 | slot,
//    key = (lo<<20)|hi (lo<hi, vertex ids < 2^20), slot = rv*6+e < 2^24
// ------------------------------------------------------------------
__global__ void k_gen_edges(const int4* __restrict__ tet, const unsigned* __restrict__ validIdx,
                            const unsigned* __restrict__ C, unsigned long long* items) {
  unsigned rv = blockIdx.x * WGS + threadIdx.x;
  if (rv >= C[0]) return;
  int4 r = tet[validIdx[rv]];
  int v[4] = {r.x, r.y, r.z, r.w};
  #pragma unroll
  for (int e = 0; e < 6; e++) {
    int a = v[c_edge[2 * e]], b = v[c_edge[2 * e + 1]];
    unsigned lo = (unsigned)min(a, b), hi = (unsigned)max(a, b);
    unsigned long long key = ((unsigned long long)lo << 20) | hi;
    items[(size_t)rv * 6 + e] = (key << 24) | (unsigned long long)(rv * 6u + e);
  }
}

__global__ void k_pad(unsigned long long* items, const unsigned* __restrict__ C, int E6) {
  int i = blockIdx.x * WGS + threadIdx.x;
  if (i >= E6) return;
  if ((unsigned)i >= 6u * C[0]) items[i] = ~0ull;  // sorts to the end every pass
}

// ------------------------------------------------------------------
// 4) LSD radix sort, 8 bits/pass on item bits [24..63] (the key).
//    ghist[d*NBLK + b]; scanned per digit row in parallel, digit bases
//    scanned separately and added at scatter time.
// ------------------------------------------------------------------
__global__ void k_hist(const unsigned long long* __restrict__ in, int n, int shift,
                       int nblk, unsigned* ghist) {
  __shared__ unsigned h[256];
  h[threadIdx.x] = 0u; __syncthreads();
  int base = blockIdx.x * 1024;
  #pragma unroll
  for (int k = 0; k < 4; k++) {
    int i = base + threadIdx.x + k * WGS;
    if (i < n) atomicAdd(&h[(unsigned)((in[i] >> shift) & 255u)], 1u);
  }
  __syncthreads();
  ghist[(size_t)threadIdx.x * nblk + blockIdx.x] = h[threadIdx.x];
}

// one block per digit: exclusive scan of its row, row total -> digitTot[d]
__global__ void k_digit_scan(unsigned* ghist, int nblk, unsigned* digitTot) {
  __shared__ unsigned wsum[NWAVE];
  __shared__ unsigned carry;
  if (threadIdx.x == 0) carry = 0u;
  __syncthreads();
  size_t row = (size_t)blockIdx.x * nblk;
  for (int base = 0; base < nblk; base += WGS) {
    int i = base + threadIdx.x;
    unsigned v = (i < nblk) ? ghist[row + i] : 0u;
    unsigned tot;
    unsigned ex = block_excl_scan_u32(v, wsum, tot);
    if (i < nblk) ghist[row + i] = carry + ex;
    __syncthreads();
    if (threadIdx.x == 0) carry += tot;
    __syncthreads();
  }
  if (threadIdx.x == 0) digitTot[blockIdx.x] = carry;
}

// single block: exclusive scan of the 256 digit totals
__global__ void k_digit_base(unsigned* digitTot) {
  __shared__ unsigned wsum[NWAVE];
  unsigned v = digitTot[threadIdx.x];
  unsigned tot;
  unsigned ex = block_excl_scan_u32(v, wsum, tot);
  digitTot[threadIdx.x] = ex;
}

__global__ void k_scatter(const unsigned long long* __restrict__ in, unsigned long long* out,
                          int n, int shift, int nblk,
                          const unsigned* __restrict__ ghist,
                          const unsigned* __restrict__ digitTot) {
  __shared__ unsigned offs[256];
  offs[threadIdx.x] = digitTot[threadIdx.x] + ghist[(size_t)threadIdx.x * nblk + blockIdx.x];
  __syncthreads();
  int base = blockIdx.x * 1024;
  #pragma unroll
  for (int k = 0; k < 4; k++) {
    int i = base + threadIdx.x + k * WGS;
    if (i < n) {
      unsigned long long it = in[i];
      unsigned d = (unsigned)((it >> shift) & 255u);
      out[atomicAdd(&offs[d], 1u)] = it;  // unstable within equal keys: duplicates are fungible
    }
  }
}

// ------------------------------------------------------------------
// 5) unique-head + surface-crossing marking, rank assignment
// ------------------------------------------------------------------
__global__ void k_hc_count(const unsigned long long* __restrict__ items, int n, int N,
                           const float* __restrict__ sdf, unsigned* blk) {
  __shared__ unsigned cnt;
  if (threadIdx.x == 0) cnt = 0u;
  __syncthreads();
  int i = blockIdx.x * WGS + threadIdx.x;
  unsigned hc = 0u;
  if (i < n) {
    unsigned long long it = items[i];
    if (it != ~0ull) {
      unsigned long long key = it >> 24;
      bool head = (i == 0) || ((items[i - 1] >> 24) != key);
      unsigned lo = (unsigned)(key >> 20), hi = (unsigned)(key & 0xFFFFFu);
      bool cross = lo < (unsigned)N && hi < (unsigned)N && ((sdf[lo] > 0.f) != (sdf[hi] > 0.f));
      hc = (head && cross) ? 1u : 0u;
    }
  }
  unsigned m = (unsigned)__ballot(hc != 0u);
  if ((threadIdx.x & 31u) == 0u && m) atomicAdd(&cnt, (unsigned)__popc(m));
  __syncthreads();
  if (threadIdx.x == 0) blk[blockIdx.x] = cnt;
}

__global__ void k_hc_emit(const unsigned long long* __restrict__ items, int n, int N,
                          const float* __restrict__ sdf, const unsigned* __restrict__ blk,
                          int* idx_map, unsigned long long* uvKey) {
  __shared__ unsigned wsum[NWAVE];
  int i = blockIdx.x * WGS + threadIdx.x;
  unsigned long long it = (i < n) ? items[i] : ~0ull;
  unsigned long long key = it >> 24;
  bool live = (i < n) && (it != ~0ull);
  bool head = false, cross = false;
  if (live) {
    head = (i == 0) || ((items[i - 1] >> 24) != key);
    unsigned lo = (unsigned)(key >> 20), hi = (unsigned)(key & 0xFFFFFu);
    cross = lo < (unsigned)N && hi < (unsigned)N && ((sdf[lo] > 0.f) != (sdf[hi] > 0.f));
  }
  unsigned hc = (head && cross) ? 1u : 0u;
  unsigned ex = block_excl_scan_flag(hc, wsum);
  if (live) {
    unsigned slot = (unsigned)(it & 0xFFFFFFull);
    if (cross) {
      unsigned id = blk[blockIdx.x] + ex + hc - 1u;  // rank of this edge's run head
      idx_map[slot] = (int)id;
      if (hc) uvKey[id] = key;
    } else {
      idx_map[slot] = -1;
    }
  }
}

// ------------------------------------------------------------------
// 6) vertex interpolation: verts = (pos_lo*(-b) + pos_hi*a) / (a-b)
// ------------------------------------------------------------------
__global__ void k_verts(const unsigned* __restrict__ C, const unsigned long long* __restrict__ uvKey,
                        const float* __restrict__ pos, const float* __restrict__ sdf,
                        float* __restrict__ out) {
  unsigned m = blockIdx.x * WGS + threadIdx.x;
  if (m >= C[4]) return;
  unsigned long long key = uvKey[m];
  unsigned lo = (unsigned)(key >> 20), hi = (unsigned)(key & 0xFFFFFu);
  float a = sdf[lo], b = sdf[hi];
  float inv = 1.0f / (a - b);
  float w0 = -b * inv, w1 = a * inv;
  #pragma unroll
  for (int j = 0; j < 3; j++)
    out[3 * (size_t)m + j] = pos[3 * (size_t)lo + j] * w0 + pos[3 * (size_t)hi + j] * w1;
}

// ------------------------------------------------------------------
// 7) faces: 1-tri tets first (tet order), then 2-tri tets; ids as f32
//    (all ids < 2^24 so exactly representable)
// ------------------------------------------------------------------
__global__ void k_faces(const unsigned* __restrict__ C, const unsigned char* __restrict__ validCase,
                        const unsigned* __restrict__ tri1, const unsigned* __restrict__ tri2,
                        const int* __restrict__ idx_map, float* __restrict__ out) {
  unsigned r = blockIdx.x * WGS + threadIdx.x;
  unsigned NF1 = C[1], NF2 = C[2], M = C[4];
  size_t base = (size_t)3 * M;
  if (r < NF1) {
    unsigned rv = tri1[r]; int cs = validCase[rv];
    #pragma unroll
    for (int j = 0; j < 3; j++)
      out[base + 3 * (size_t)r + j] = (float)idx_map[rv * 6u + c_tri[cs][j]];
  }
  if (r < NF2) {
    unsigned rv = tri2[r]; int cs = validCase[rv];
    size_t b2 = base + (size_t)3 * NF1 + (size_t)6 * r;
    #pragma unroll
    for (int j = 0; j < 6; j++)
      out[b2 + j] = (float)idx_map[rv * 6u + c_tri[cs][j]];
  }
}

// =====================================================================
extern "C" void kernel_launch(void* const* d_in, const int* in_sizes, int n_in,
                              void* d_out, int out_size, void* d_ws, size_t ws_size,
                              hipStream_t stream) {
  const float* pos = (const float*)d_in[0];
  const float* sdf = (const float*)d_in[1];
  const int4*  tet = (const int4*)d_in[2];   // int32 per harness convention, 4 per tet
  float* out = (float*)d_out;

  const int N    = in_sizes[1];
  const int F    = in_sizes[2] / 4;
  const int E6   = 6 * F;
  const int NB1  = (F + WGS - 1) / WGS;
  const int GB6  = (E6 + WGS - 1) / WGS;
  const int NBLK = (E6 + 1023) / 1024;
  const int NBINS = 256 * NBLK;

  char* basep = (char*)d_ws;
  size_t off = 0;
  auto carve = [&](size_t bytes) -> void* {
    void* r = basep + off;
    off = (off + bytes + 255) & ~(size_t)255;
    return r;
  };
  unsigned*            C         = (unsigned*)carve(64);
  unsigned*            blkV      = (unsigned*)carve((size_t)NB1 * 4);
  unsigned*            blk1      = (unsigned*)carve((size_t)NB1 * 4);
  unsigned*            blk2      = (unsigned*)carve((size_t)NB1 * 4);
  unsigned char*       caseArr   = (unsigned char*)carve((size_t)F);
  unsigned*            validIdx  = (unsigned*)carve((size_t)F * 4);
  unsigned char*       validCase = (unsigned char*)carve((size_t)F);
  unsigned*            tri1      = (unsigned*)carve((size_t)F * 4);
  unsigned*            tri2      = (unsigned*)carve((size_t)F * 4);
  unsigned long long*  itemsA    = (unsigned long long*)carve((size_t)E6 * 8);
  unsigned long long*  itemsB    = (unsigned long long*)carve((size_t)E6 * 8);
  unsigned*            ghist     = (unsigned*)carve((size_t)NBINS * 4);
  unsigned*            digitTot  = (unsigned*)carve(256 * 4);
  unsigned*            hcBlk     = (unsigned*)carve((size_t)GB6 * 4);
  int*                 idx_map   = (int*)carve((size_t)E6 * 4);
  unsigned long long*  uvKey     = (unsigned long long*)carve((size_t)E6 * 8);
  (void)n_in; (void)out_size; (void)ws_size;

  k_classify<<<NB1, WGS, 0, stream>>>(tet, sdf, F, blkV, blk1, blk2, caseArr);
  k_scan_arr<<<1, WGS, 0, stream>>>(blkV, NB1, &C[0]);
  k_scan_arr<<<1, WGS, 0, stream>>>(blk1, NB1, &C[1]);
  k_scan_arr<<<1, WGS, 0, stream>>>(blk2, NB1, &C[2]);
  k_compact<<<NB1, WGS, 0, stream>>>(caseArr, F, blkV, blk1, blk2,
                                     validIdx, validCase, tri1, tri2);
  k_gen_edges<<<NB1, WGS, 0, stream>>>(tet, validIdx, C, itemsA);
  k_pad<<<GB6, WGS, 0, stream>>>(itemsA, C, E6);

  unsigned long long* src = itemsA;
  unsigned long long* dst = itemsB;
  for (int p = 0; p < 5; p++) {               // key bits 24..63, 8 bits/pass
    int shift = 24 + 8 * p;
    k_hist<<<NBLK, WGS, 0, stream>>>(src, E6, shift, NBLK, ghist);
    k_digit_scan<<<256, WGS, 0, stream>>>(ghist, NBLK, digitTot);
    k_digit_base<<<1, WGS, 0, stream>>>(digitTot);
    k_scatter<<<NBLK, WGS, 0, stream>>>(src, dst, E6, shift, NBLK, ghist, digitTot);
    unsigned long long* t = src; src = dst; dst = t;
  }
  // sorted items now in `src`

  k_hc_count<<<GB6, WGS, 0, stream>>>(src, E6, N, sdf, hcBlk);
  k_scan_arr<<<1, WGS, 0, stream>>>(hcBlk, GB6, &C[4]);
  k_hc_emit<<<GB6, WGS, 0, stream>>>(src, E6, N, sdf, hcBlk, idx_map, uvKey);
  k_verts<<<GB6, WGS, 0, stream>>>(C, uvKey, pos, sdf, out);
  k_faces<<<NB1, WGS, 0, stream>>>(C, validCase, tri1, tri2, idx_map, out);
}